// LSTM_87978110091856
// MI455X (gfx1250) — compile-verified
//
#include <hip/hip_runtime.h>
#include <hip/hip_bf16.h>

// CDNA5 / gfx1250 fused LSTM:
//  - one wave (32 lanes) per 16 batch rows, full T-loop per wave
//  - z = [x_t, h] @ W + b' via V_WMMA_F32_16X16X4_F32 (84 wmma / step)
//  - Wx packed in LDS in B-operand layout, Wh cached in VGPRs
//  - gate-interleaved column permutation so gate reads are float4
//  - single-wave workgroups: LDS in-order semantics, no barriers needed

typedef __attribute__((ext_vector_type(2))) float v2f;
typedef __attribute__((ext_vector_type(8))) float v8f;

#define T_STEPS 512
#define D_IN    32
#define H_HID   24
#define C_OUT   10
#define NB4H    96      // 4H
#define NKB_X   8       // D/4 k-blocks from x
#define NKB_H   6       // H/4 k-blocks from h
#define NNT     6       // 96/16 N tiles
#define ZSTRIDE 100     // padded row stride (floats) for z scratch, 16B-aligned rows
#define HSTRIDE 26      // padded row stride (floats) for h buffer, 8B-aligned pairs

__device__ __forceinline__ float fsig(float x) {
    return __builtin_amdgcn_rcpf(1.0f + __expf(-x));
}
__device__ __forceinline__ float ftanh_f(float x) {
    return 2.0f * __builtin_amdgcn_rcpf(1.0f + __expf(-2.0f * x)) - 1.0f;
}

__global__ __launch_bounds__(32) void lstm_wmma_f32_kernel(
    const float* __restrict__ x,     // [B, T, D]
    const float* __restrict__ Wl,    // [D+H, 4H] gate order i,j,f,o
    const float* __restrict__ bl,    // [4H]
    const float* __restrict__ Wo,    // [H, C]
    const float* __restrict__ bo,    // [C]
    float* __restrict__ out)         // [B, C]
{
    __shared__ float wpk[NKB_X * NNT * 32 * 2];   // packed Wx B-operands (float2/lane)
    __shared__ float zscr[16 * ZSTRIDE];          // z spill, columns in N' = 4g+gate order
    __shared__ float hbuf[16 * HSTRIDE];          // h state, A-operand readable layout

    const int L  = threadIdx.x;   // 0..31
    const int m  = L & 15;        // row within 16x16 tile (A: M, C: N-local)
    const int hi = L >> 4;        // half-wave select
    const int b0 = blockIdx.x * 16;

    // ---- one-time: pack Wx into LDS B-operand layout, gate-interleaved columns ----
    // B layout (mirrors ISA A 16x4 layout): VGPR0 = B[K=4kb+2*hi][N], VGPR1 = K+1
    // column permutation: N' = 4*g + gate  <->  original col = gate*24 + g
    #pragma unroll 1
    for (int idx = 0; idx < NKB_X * NNT; ++idx) {
        const int kb = idx / NNT, nt = idx % NNT;
        const int Np = nt * 16 + m;
        const int gate = Np & 3, g = Np >> 2;
        const int c = gate * H_HID + g;
        const int r = kb * 4 + 2 * hi;
        float2 v = make_float2(Wl[r * NB4H + c], Wl[(r + 1) * NB4H + c]);
        ((float2*)wpk)[idx * 32 + L] = v;
    }

    // ---- Wh B-operands cached in registers (recurrent critical path) ----
    v2f wh[NKB_H][NNT];
    #pragma unroll
    for (int kbh = 0; kbh < NKB_H; ++kbh) {
        #pragma unroll
        for (int nt = 0; nt < NNT; ++nt) {
            const int Np = nt * 16 + m;
            const int gate = Np & 3, g = Np >> 2;
            const int c = gate * H_HID + g;
            const int r = (NKB_X + kbh) * 4 + 2 * hi;
            v2f w; w.x = Wl[r * NB4H + c]; w.y = Wl[(r + 1) * NB4H + c];
            wh[kbh][nt] = w;
        }
    }

    // ---- bias per N' column (forget_bias = 1.0 folded in) ----
    float bcol[NNT];
    #pragma unroll
    for (int nt = 0; nt < NNT; ++nt) {
        const int Np = nt * 16 + m;
        const int gate = Np & 3, g = Np >> 2;
        bcol[nt] = bl[gate * H_HID + g] + (gate == 2 ? 1.0f : 0.0f);
    }

    // ---- init state ----
    for (int i = L; i < 16 * HSTRIDE; i += 32) hbuf[i] = 0.0f;
    float cc[12];
    #pragma unroll
    for (int j = 0; j < 12; ++j) cc[j] = 0.0f;

    const float* rowp = x + (size_t)(b0 + m) * T_STEPS * D_IN;

    // ================= time loop =================
    #pragma unroll 1
    for (int t = 0; t < T_STEPS; ++t) {
        // prefetch next timestep's 128B row chunk
        if (t + 1 < T_STEPS)
            __builtin_prefetch(rowp + (size_t)(t + 1) * D_IN + 2 * hi, 0, 0);

        // A-operands from x_t: lane L -> x[row m][4kb + 2*hi (+1)]
        const float* xt = rowp + (size_t)t * D_IN;
        v2f ax[NKB_X];
        #pragma unroll
        for (int kb = 0; kb < NKB_X; ++kb) {
            float2 v = *(const float2*)(xt + kb * 4 + 2 * hi);
            v2f a; a.x = v.x; a.y = v.y;
            ax[kb] = a;
        }

        // accumulators = bias
        v8f acc[NNT];
        #pragma unroll
        for (int nt = 0; nt < NNT; ++nt) {
            v8f z;
            #pragma unroll
            for (int v = 0; v < 8; ++v) z[v] = bcol[nt];
            acc[nt] = z;
        }

        // x @ Wx : 8 k-blocks x 6 N tiles (6 independent accumulation chains)
        #pragma unroll
        for (int kb = 0; kb < NKB_X; ++kb) {
            #pragma unroll
            for (int nt = 0; nt < NNT; ++nt) {
                float2 bw = ((const float2*)wpk)[(kb * NNT + nt) * 32 + L];
                v2f b; b.x = bw.x; b.y = bw.y;
                acc[nt] = __builtin_amdgcn_wmma_f32_16x16x4_f32(
                    false, ax[kb], false, b, (short)0, acc[nt], false, false);
            }
        }

        // h @ Wh : 6 k-blocks x 6 N tiles, Wh from registers
        #pragma unroll
        for (int kbh = 0; kbh < NKB_H; ++kbh) {
            float2 hv = *(const float2*)(&hbuf[m * HSTRIDE + kbh * 4 + 2 * hi]);
            v2f ah; ah.x = hv.x; ah.y = hv.y;
            #pragma unroll
            for (int nt = 0; nt < NNT; ++nt) {
                acc[nt] = __builtin_amdgcn_wmma_f32_16x16x4_f32(
                    false, ah, false, wh[kbh][nt], (short)0, acc[nt], false, false);
            }
        }

        // spill z (C layout: VGPR v -> row v+8*hi, col = nt*16 + m) to LDS
        #pragma unroll
        for (int nt = 0; nt < NNT; ++nt) {
            #pragma unroll
            for (int v = 0; v < 8; ++v)
                zscr[(v + 8 * hi) * ZSTRIDE + nt * 16 + m] = acc[nt][v];
        }

        // gates: this lane owns (M=m, g = 12*hi + j); columns are (i,j,f,o) packed
        float hval[12];
        #pragma unroll
        for (int j = 0; j < 12; ++j) {
            const int g = 12 * hi + j;
            float4 z4 = *(const float4*)(&zscr[m * ZSTRIDE + 4 * g]);
            float gi = fsig(z4.x);
            float gj = ftanh_f(z4.y);
            float gf = fsig(z4.z);           // +1.0 already folded into bias
            float go = fsig(z4.w);
            float cn = gf * cc[j] + gi * gj;
            cc[j] = cn;
            hval[j] = go * ftanh_f(cn);
        }

        // write h back in A-readable layout: hbuf[m][12*hi + j]
        #pragma unroll
        for (int j = 0; j < 6; ++j) {
            *(float2*)(&hbuf[m * HSTRIDE + 12 * hi + 2 * j]) =
                make_float2(hval[2 * j], hval[2 * j + 1]);
        }
    }

    // ================= head: softmax(h @ Wo + bo), lanes 0..15 =================
    if (hi == 0) {
        float hrow[H_HID];
        #pragma unroll
        for (int k = 0; k < H_HID; ++k) hrow[k] = hbuf[m * HSTRIDE + k];

        float logit[C_OUT];
        #pragma unroll
        for (int c = 0; c < C_OUT; ++c) {
            float s = bo[c];
            #pragma unroll
            for (int k = 0; k < H_HID; ++k) s += hrow[k] * Wo[k * C_OUT + c];
            logit[c] = s;
        }
        float mx = logit[0];
        #pragma unroll
        for (int c = 1; c < C_OUT; ++c) mx = fmaxf(mx, logit[c]);
        float sum = 0.0f;
        #pragma unroll
        for (int c = 0; c < C_OUT; ++c) { float e = __expf(logit[c] - mx); logit[c] = e; sum += e; }
        float inv = __builtin_amdgcn_rcpf(sum);
        #pragma unroll
        for (int c = 0; c < C_OUT; ++c)
            out[(size_t)(b0 + m) * C_OUT + c] = logit[c] * inv;
    }
}

extern "C" void kernel_launch(void* const* d_in, const int* in_sizes, int n_in,
                              void* d_out, int out_size, void* d_ws, size_t ws_size,
                              hipStream_t stream) {
    (void)n_in; (void)out_size; (void)d_ws; (void)ws_size;
    const float* x  = (const float*)d_in[0];   // [B, T, D]
    const float* Wl = (const float*)d_in[1];   // [D+H, 4H]
    const float* bl = (const float*)d_in[2];   // [4H]
    const float* Wo = (const float*)d_in[3];   // [H, C]
    const float* bo = (const float*)d_in[4];   // [C]
    float* out = (float*)d_out;                // [B, C]

    const int B = in_sizes[0] / (T_STEPS * D_IN);   // 2048
    dim3 grid(B / 16), block(32);
    lstm_wmma_f32_kernel<<<grid, block, 0, stream>>>(x, Wl, bl, Wo, bo, out);
}